// AttnResBase_87582973100691
// MI455X (gfx1250) — compile-verified
//
#include <hip/hip_runtime.h>
#include <stdint.h>

// Problem constants (reference: B=4, S=4096, H=768, L=8)
#define HDIM 768
#define LNUM 8
#define TPOS 4      // positions per block
#define BLOCK 256   // 8 waves of 32
#define F4_PER_VEC (HDIM / 4)        // 192 float4 per 768-float vector
#define F4_ITERS   (HDIM / (32 * 4)) // 6 b128 wave-iterations per vector

// ---- CDNA5 async global->LDS helpers (gfx1250) -----------------------------

// Raw 32-bit LDS byte address of a __shared__ object (AS3 pointers are the
// LDS offset; LDS_BASE is added by hardware).
__device__ __forceinline__ uint32_t lds_addr_of(const void* p) {
  return (uint32_t)(size_t)(const __attribute__((address_space(3))) void*)p;
}

// GLOBAL_LOAD_ASYNC_TO_LDS_B128: each lane copies 16B from global vaddr to LDS at vdst.
// Tracked by ASYNCcnt, not LOADcnt.
__device__ __forceinline__ void async_ld_b128(uint32_t lds_byte_addr,
                                              const void* gaddr) {
  asm volatile("global_load_async_to_lds_b128 %0, %1, off"
               :
               : "v"(lds_byte_addr), "v"(gaddr)
               : "memory");
}

__device__ __forceinline__ void wait_async_zero() {
  asm volatile("s_wait_asynccnt 0" ::: "memory");
}

// ---- Kernel 1: qW = (q @ W_key) / sqrt(H) ---------------------------------
// qW[h] = sum_j q[j] * W[j*H + h]  (W is [out,in] row-major; q @ W indexes
// W[j, h] = W[j*H + h]). Coalesced across h.
__global__ __launch_bounds__(256) void qw_project_kernel(
    const float* __restrict__ W, const float* __restrict__ q,
    float* __restrict__ qW) {
  __shared__ float qs[HDIM];
  const int tid = threadIdx.x;
  qs[tid]       = q[tid];
  qs[tid + 256] = q[tid + 256];
  qs[tid + 512] = q[tid + 512];
  __syncthreads();
  const int h = blockIdx.x * 256 + tid;
  float acc = 0.f;
  #pragma unroll 8
  for (int j = 0; j < HDIM; ++j) acc += qs[j] * W[(size_t)j * HDIM + h];
  qW[h] = acc * 0.03608439182435161f;  // 1/sqrt(768)
}

// ---- Kernel 2: single-pass attention over preceding layers ----------------
// One block = TPOS consecutive positions. preceding index ((l*B+b)*S+s)*H
// collapses to (l*N + pos)*H with N = B*S, pos = b*S + s.
__global__ __launch_bounds__(BLOCK) void attn_res_kernel(
    const float* __restrict__ preceding,  // [L, N, H]
    const float* __restrict__ qW,         // [H], pre-scaled by 1/sqrt(H)
    float* __restrict__ out,              // [N, H]
    int N) {
  __shared__ float buf[TPOS][LNUM][HDIM];     // 96 KB tile
  __shared__ float qws[HDIM];                 // 3 KB
  __shared__ float scores[TPOS * LNUM];       // 32 dots

  const int tid  = threadIdx.x;
  const int wave = tid >> 5;   // 0..7  -> layer for the async stage
  const int lane = tid & 31;
  const long pos0 = (long)blockIdx.x * TPOS;

  // Stage 1: async-copy this block's tile (TPOS x LNUM x 768 f32) into LDS.
  // Wave w streams layer w: TPOS positions x 6 b128 wave-ops = 24 async ops.
  {
    const int l = wave;
    #pragma unroll
    for (int t = 0; t < TPOS; ++t) {
      const float* src = preceding + ((long)l * N + (pos0 + t)) * HDIM;
      const uint32_t dst = lds_addr_of(&buf[t][l][0]);
      #pragma unroll
      for (int i = 0; i < F4_ITERS; ++i) {
        const int f = i * 32 + lane;                 // float4 index 0..191
        async_ld_b128(dst + (uint32_t)f * 16u, src + f * 4);
      }
    }
  }
  // Overlap: pull qW into LDS with normal loads while async copies fly.
  qws[tid]       = qW[tid];
  qws[tid + 256] = qW[tid + 256];
  qws[tid + 512] = qW[tid + 512];

  wait_async_zero();   // this wave's async writes visible in LDS
  __syncthreads();     // all waves' writes visible

  // Stage 2: 32 dot products of length 768; wave w owns d = 4w..4w+3
  // (d = t*8 + l). qW slice for this lane cached in 24 VGPRs and reused.
  float qv[24];
  #pragma unroll
  for (int k = 0; k < 24; ++k) qv[k] = qws[lane + 32 * k];

  #pragma unroll
  for (int dd = 0; dd < 4; ++dd) {
    const int d = wave * 4 + dd;
    const float* v = &buf[0][0][0] + d * HDIM;
    float p = 0.f;
    #pragma unroll
    for (int k = 0; k < 24; ++k) p += v[lane + 32 * k] * qv[k];
    #pragma unroll
    for (int off = 16; off > 0; off >>= 1) p += __shfl_xor(p, off, 32);
    if (lane == 0) scores[d] = p;
  }
  __syncthreads();

  // Stage 3: softmax over L=8 (per position) + weighted sum + store.
  // 64 threads per position; each thread emits 3 coalesced float4.
  const int t = tid >> 6;        // 0..3
  const int fbase = tid & 63;    // float4 lane within position
  float w[LNUM];
  {
    float m = scores[t * LNUM];
    #pragma unroll
    for (int l = 1; l < LNUM; ++l) m = fmaxf(m, scores[t * LNUM + l]);
    float sum = 0.f;
    #pragma unroll
    for (int l = 0; l < LNUM; ++l) {
      w[l] = __expf(scores[t * LNUM + l] - m);
      sum += w[l];
    }
    const float inv = 1.f / sum;
    #pragma unroll
    for (int l = 0; l < LNUM; ++l) w[l] *= inv;
  }

  float4* outp = (float4*)(out + (pos0 + t) * HDIM);
  const float4* bufv = (const float4*)&buf[t][0][0];  // [LNUM][192]
  #pragma unroll
  for (int j = 0; j < 3; ++j) {
    const int f = fbase + j * 64;   // 0..191
    float4 acc = make_float4(0.f, 0.f, 0.f, 0.f);
    #pragma unroll
    for (int l = 0; l < LNUM; ++l) {
      const float4 v = bufv[l * F4_PER_VEC + f];
      acc.x += w[l] * v.x;
      acc.y += w[l] * v.y;
      acc.z += w[l] * v.z;
      acc.w += w[l] * v.w;
    }
    outp[f] = acc;
  }
}

// ---- Launch ---------------------------------------------------------------
extern "C" void kernel_launch(void* const* d_in, const int* in_sizes, int n_in,
                              void* d_out, int out_size, void* d_ws,
                              size_t ws_size, hipStream_t stream) {
  // setup_inputs order: current_output (unused), preceding, W_key, query
  const float* preceding = (const float*)d_in[1];
  const float* W_key     = (const float*)d_in[2];
  const float* query     = (const float*)d_in[3];
  float* out = (float*)d_out;
  float* qW  = (float*)d_ws;  // 768 floats of scratch

  const int N = in_sizes[0] / HDIM;  // B*S flat positions (16384)

  qw_project_kernel<<<HDIM / 256, 256, 0, stream>>>(W_key, query, qW);
  attn_res_kernel<<<N / TPOS, BLOCK, 0, stream>>>(preceding, qW, out, N);
}